// RoiPoolingConv_16492674416880
// MI455X (gfx1250) — compile-verified
//
#include <hip/hip_runtime.h>
#include <stdint.h>

// ROI-align 7x7, (1,256,256,512) fp32 NHWC, 300 ROIs (x,y,w,h int32).
// CDNA5 async-pipelined gather: taps are DMA'd global->LDS with
// GLOBAL_LOAD_ASYNC_TO_LDS_B128 (ASYNCcnt-tracked), triple-buffered,
// depth-2 pipeline with partial s_wait_asynccnt waits. Blend in VALU,
// non-temporal B128 stores (write-once output keeps L2 for the feature map).

typedef __attribute__((ext_vector_type(4))) float v4f;
typedef __attribute__((ext_vector_type(4))) int   v4i;

// Builtin prototype (from hipcc diagnostic): arg0 = v4i in addrspace(1),
// arg1 = LDS side, then (offset, cpol) immediates.
typedef __attribute__((address_space(1))) v4i gas_v4i;   // global
typedef __attribute__((address_space(3))) v4i las_v4i;   // LDS

#define POOL_SIZE 7
#define NUM_ROIS  300
#define IMG_W     256
#define IMG_C     512
#define CV        (IMG_C / 4)   // float4s per pixel = 128
#define BLOCK     128           // 4 x wave32; thread t <-> float4 channel t
#define NBUF      3             // triple buffer (no LDS WAR hazard)

// ---- CDNA5 async global->LDS 16B copy, tracked by ASYNCcnt ----
__device__ __forceinline__ void async_copy16(const float* g, v4f* l) {
#if __has_builtin(__builtin_amdgcn_global_load_async_to_lds_b128)
    // Int-detour casts: inttoptr is always legal. AS1 ptr value = global VA;
    // AS3 ptr value = low 32 bits of the generic pointer, which per the flat
    // aperture mapping is the DS-visible LDS offset.
    gas_v4i* gp = (gas_v4i*)(uintptr_t)g;
    las_v4i* lp = (las_v4i*)(unsigned)(uintptr_t)l;
    __builtin_amdgcn_global_load_async_to_lds_b128(gp, lp, 0 /*offset*/, 0 /*cpol*/);
#else
    // VDST = per-lane LDS byte address, VADDR = per-lane 64-bit global address
    asm volatile("global_load_async_to_lds_b128 %0, %1, off"
                 :: "v"((unsigned)(uintptr_t)l), "v"(g)
                 : "memory");
#endif
}

template <int N>
__device__ __forceinline__ void wait_async() {
#if __has_builtin(__builtin_amdgcn_s_wait_asynccnt)
    __builtin_amdgcn_s_wait_asynccnt(N);
#else
    asm volatile("s_wait_asynccnt %0" :: "i"(N) : "memory");
#endif
}

__global__ __launch_bounds__(BLOCK) void roi_align_7x7_kernel(
    const float* __restrict__ img,   // (256,256,512)
    const int*   __restrict__ rois,  // (300,4): x,y,w,h
    float*       __restrict__ out)   // (300,7,7,512)
{
    const int py  = blockIdx.x;      // 0..6
    const int roi = blockIdx.y;      // 0..299
    const int t   = threadIdx.x;     // 0..127 -> channels 4t..4t+3

    // Staging: [buffer][tap][float4 channel chunk]; waves touch disjoint [t]
    // slots and each wave waits only on its own ASYNCcnt -> no barriers.
    __shared__ v4f stage[NBUF][4][BLOCK];   // 24 KB

    const int rx = rois[roi * 4 + 0];
    const int ry = rois[roi * 4 + 1];
    const int rw = rois[roi * 4 + 2];
    const int rh = rois[roi * 4 + 3];

    // ---- y axis (uniform -> scalar float pipe) ----
    float sy = ((float)py + 0.5f) * ((float)rh * (1.0f / POOL_SIZE)) - 0.5f;
    sy = fminf(fmaxf(sy, 0.0f), (float)(rh - 1));
    const int   iy0 = (int)floorf(sy);
    const int   iy1 = min(iy0 + 1, rh - 1);
    const float fy  = sy - (float)iy0;
    const float gy  = 1.0f - fy;

    const v4f* __restrict__ imgv = (const v4f*)img;
    const long rowA = (long)(ry + iy0) * IMG_W;
    const long rowB = (long)(ry + iy1) * IMG_W;

    // ---- x axis: all 7 sample positions (uniform) ----
    int   x0[POOL_SIZE], x1[POOL_SIZE];
    float fx[POOL_SIZE];
    const float xs = (float)rw * (1.0f / POOL_SIZE);
#pragma unroll
    for (int px = 0; px < POOL_SIZE; ++px) {
        float sx = ((float)px + 0.5f) * xs - 0.5f;
        sx = fminf(fmaxf(sx, 0.0f), (float)(rw - 1));
        const int i0 = (int)floorf(sx);
        x0[px] = rx + i0;
        x1[px] = rx + min(i0 + 1, rw - 1);
        fx[px] = sx - (float)i0;
    }

    // Issue 4 async tap copies (2 rows x 2 cols) for sample px into buffer b.
    auto issue = [&](int px, int b) {
        async_copy16((const float*)&imgv[(rowA + x0[px]) * CV + t], &stage[b][0][t]);
        async_copy16((const float*)&imgv[(rowA + x1[px]) * CV + t], &stage[b][1][t]);
        async_copy16((const float*)&imgv[(rowB + x0[px]) * CV + t], &stage[b][2][t]);
        async_copy16((const float*)&imgv[(rowB + x1[px]) * CV + t], &stage[b][3][t]);
    };

    v4f* outv = (v4f*)out;
    const long outBase = ((long)(roi * POOL_SIZE + py) * POOL_SIZE) * CV + t;

    // Depth-2 software pipeline over the 7 x-samples.
    issue(0, 0);
    issue(1, 1);

#pragma unroll
    for (int px = 0; px < POOL_SIZE; ++px) {
        if (px + 2 < POOL_SIZE) {
            issue(px + 2, (px + 2) % NBUF);
            wait_async<8>();        // 12 outstanding -> oldest 4 (px) done
        } else if (px + 1 < POOL_SIZE) {
            wait_async<4>();        // 8 outstanding  -> oldest 4 (px) done
        } else {
            wait_async<0>();        // last group
        }

        const int b = px % NBUF;
        const v4f a00 = stage[b][0][t];
        const v4f a01 = stage[b][1][t];
        const v4f a10 = stage[b][2][t];
        const v4f a11 = stage[b][3][t];

        const float wx = fx[px];
        const float gx = 1.0f - wx;

        const v4f r = (a00 * gx + a01 * wx) * gy
                    + (a10 * gx + a11 * wx) * fy;

        // Write-once output: non-temporal store keeps L2 for the feature map.
        __builtin_nontemporal_store(r, &outv[outBase + (long)px * CV]);
    }
}

extern "C" void kernel_launch(void* const* d_in, const int* in_sizes, int n_in,
                              void* d_out, int out_size, void* d_ws, size_t ws_size,
                              hipStream_t stream) {
    (void)in_sizes; (void)n_in; (void)out_size; (void)d_ws; (void)ws_size;
    const float* img  = (const float*)d_in[0];
    const int*   rois = (const int*)d_in[1];
    float*       out  = (float*)d_out;

    dim3 grid(POOL_SIZE, NUM_ROIS);   // (py, roi) -> 2100 blocks
    dim3 block(BLOCK);                // 4 x wave32
    roi_align_7x7_kernel<<<grid, block, 0, stream>>>(img, rois, out);
}